// ReasoningByteEmbeddings_9998683865347
// MI455X (gfx1250) — compile-verified
//
#include <hip/hip_runtime.h>
#include <stdint.h>

// ReasoningByteEmbeddings: n-gram hash embedding gather-sum.
// tokens: (32,2048) i32 in [0,256); tables: (6,500000,128) f32; out: (32,2048,128) f32.
// Memory-bound gather (~235 MB/launch -> ~10us floor at 23.3 TB/s on MI455X).
// Two-phase plan keeps instruction issue far below the memory floor:
//   K1: per-LANE rolling-hash index build (VALU-parallel), 8 indices/position -> ws.
//   K2: per-WAVE CDNA5 TDM gather (tensor_load_to_lds, gather_mode, 32-bit indices,
//       6 rows x 512B per op) into LDS, float4 reduce, non-temporal b128 store.

#define NGRAMS 6
#define VOCAB  500000
#define DIM    128
#define BATCHN 32
#define SEQN   2048
#define HBASE  257
#define WAVES_PER_BLOCK 8
#define POSITIONS (BATCHN * SEQN)          // 65536

typedef float v4f  __attribute__((ext_vector_type(4)));
typedef unsigned int u32x4 __attribute__((ext_vector_type(4)));
typedef int i32x4 __attribute__((ext_vector_type(4)));
typedef int i32x8 __attribute__((ext_vector_type(8)));

// ---------------------------------------------------------------------------
// Shared helper: rolling polynomial hash -> 6 stacked-table row indices.
// h_{n+1} = (h_n * 257 + tok[i-n]) % V ; positions with s < n-1 use raw byte.
// Row index for ngram k is k*VOCAB + idx_k  (stacked (6*500000, 128) view).
// ---------------------------------------------------------------------------
__device__ __forceinline__ void ngram_row_indices(const int* __restrict__ trow,
                                                  int s, int r[8])
{
    int toks[8];
#pragma unroll
    for (int j = 0; j < 8; ++j) {
        const int sj = s - j;
        const int t  = trow[(sj < 0) ? 0 : sj];   // clamped, always in-bounds
        toks[j] = (sj < 0) ? 0 : t;
    }
    const int x0 = toks[0];
    int h = x0;
#pragma unroll
    for (int j = 1; j <= 7; ++j) {
        h = (h * HBASE + toks[j]) % VOCAB;        // < 257*500000+255 < 2^31
        if (j >= 2) {
            const int k = j - 2;
            r[k] = k * VOCAB + ((s >= j) ? h : x0);
        }
    }
    r[6] = r[5];                                   // pad slots (TDM uses 6 valid)
    r[7] = r[5];
}

// ---------------------------------------------------------------------------
// K1: one position per lane; fully VALU-parallel hash; 2x b128 stores to ws.
// ---------------------------------------------------------------------------
__global__ __launch_bounds__(256) void ngram_hash_kernel(
    const int* __restrict__ tokens,
    int* __restrict__ rws)
{
    const int pos = blockIdx.x * 256 + threadIdx.x;        // grid exact: 65536
    const int b = pos >> 11;
    const int s = pos & (SEQN - 1);

    int r[8];
    ngram_row_indices(tokens + (b << 11), s, r);

    i32x4* o = (i32x4*)(rws + ((size_t)pos << 3));         // 32B aligned
    i32x4 lo; lo[0] = r[0]; lo[1] = r[1]; lo[2] = r[2]; lo[3] = r[3];
    i32x4 hi; hi[0] = r[4]; hi[1] = r[5]; hi[2] = r[6]; hi[3] = r[7];
    o[0] = lo;
    o[1] = hi;
}

// ---------------------------------------------------------------------------
// K2: one position per wave; scalar-load 8 indices; one TDM gather op fetches
// all 6 embedding rows (3 KB) into LDS; lanes reduce float4 slices; NT store.
// ---------------------------------------------------------------------------
__global__ __launch_bounds__(256) void ngram_gather_kernel(
    const int* __restrict__ rws,
    const float* __restrict__ tables,
    float* __restrict__ out)
{
    __shared__ float sh[WAVES_PER_BLOCK * NGRAMS * DIM];   // 24 KB

    const int lane = threadIdx.x & 31;
    const int wv   = threadIdx.x >> 5;
    const int pos  = __builtin_amdgcn_readfirstlane(blockIdx.x * WAVES_PER_BLOCK + wv);

    // Uniform address -> SMEM (s_load_b256) of the 8 precomputed row indices.
    const int* rp = rws + ((size_t)pos << 3);
    const int r0 = rp[0], r1 = rp[1], r2 = rp[2], r3 = rp[3];
    const int r4 = rp[4], r5 = rp[5], r6 = rp[6], r7 = rp[7];

    float* myLds = sh + wv * (NGRAMS * DIM);
    v4f acc = (v4f){0.f, 0.f, 0.f, 0.f};

#if __has_builtin(__builtin_amdgcn_tensor_load_to_lds) && __has_builtin(__builtin_amdgcn_s_wait_tensorcnt)
    const unsigned lds_addr =
        __builtin_amdgcn_readfirstlane((unsigned)(size_t)myLds);  // low 32b = LDS offset
    const unsigned long long ga = (unsigned long long)(const void*)tables;

    // D# group 0: count=1 (valid), gather_mode=1, 32-bit indices; lds_addr;
    // global_addr[56:0]; type=2.
    u32x4 g0;
    g0[0] = 0xC0000001u;
    g0[1] = lds_addr;
    g0[2] = (unsigned)ga;
    g0[3] = ((unsigned)(ga >> 32) & 0x01FFFFFFu) | 0x80000000u;

    // D# group 1: wg_mask=0, data_size=4B; tensor_dim0=128; tensor_dim1=3,000,000;
    // tile_dim0=128; tile_dim1=6 valid indices; tensor_dim0_stride=128 elements.
    const unsigned NROWS = (unsigned)(NGRAMS * VOCAB);   // 0x2DC6C0
    i32x8 g1;
    g1[0] = 0x00020000;                                   // data_size=2 (4 bytes)
    g1[1] = (int)((unsigned)DIM << 16);                   // tensor_dim0 lo16 @ [63:48]
    g1[2] = (int)((NROWS & 0xFFFFu) << 16);               // dim0 hi16=0 | tensor_dim1 lo16
    g1[3] = (int)((NROWS >> 16) | ((unsigned)DIM << 16)); // tensor_dim1 hi16 | tile_dim0
    g1[4] = NGRAMS;                                       // tile_dim1 = #valid indices
    g1[5] = DIM;                                          // tensor_dim0_stride lo32
    g1[6] = 0;
    g1[7] = 0;

    // Groups 2/3: the 8 row indices (slots 6,7 are harmless pad copies).
    i32x4 g2, g3;
    g2[0] = r0; g2[1] = r1; g2[2] = r2; g2[3] = r3;
    g3[0] = r4; g3[1] = r5; g3[2] = r6; g3[3] = r7;

    // 6-arg form (clang-23 / therock-10.0): extra i32x8 = unused VADDR4 slot.
    i32x8 g4;
    g4[0]=0; g4[1]=0; g4[2]=0; g4[3]=0; g4[4]=0; g4[5]=0; g4[6]=0; g4[7]=0;

    __builtin_amdgcn_tensor_load_to_lds(g0, g1, g2, g3, g4, 0);
    __builtin_amdgcn_s_wait_tensorcnt(0);
    __asm__ volatile("" ::: "memory");

#pragma unroll
    for (int k = 0; k < NGRAMS; ++k) {
        const v4f* p = (const v4f*)(myLds + k * DIM);
        acc += p[lane];                                   // ds_load_b128
    }
#else
    // ---- fallback: direct b128 gathers with NT hint ----
    const int rr[NGRAMS] = {r0, r1, r2, r3, r4, r5};
#pragma unroll
    for (int k = 0; k < NGRAMS; ++k) {
        const v4f* row = (const v4f*)(tables + (size_t)rr[k] * DIM);
        acc += __builtin_nontemporal_load(&row[lane]);
    }
#endif

    acc *= (1.0f / (float)NGRAMS);
    v4f* o = (v4f*)(out + (size_t)pos * DIM);
    __builtin_nontemporal_store(acc, &o[lane]);           // streaming 33.5 MB output
}

// ---------------------------------------------------------------------------
// Fused single-kernel fallback (used only if workspace is too small).
// ---------------------------------------------------------------------------
__global__ __launch_bounds__(256) void ngram_fused_kernel(
    const int* __restrict__ tokens,
    const float* __restrict__ tables,
    float* __restrict__ out)
{
    const int lane = threadIdx.x & 31;
    const int wv   = threadIdx.x >> 5;
    const int pos  = __builtin_amdgcn_readfirstlane(blockIdx.x * WAVES_PER_BLOCK + wv);
    const int b = pos >> 11;
    const int s = pos & (SEQN - 1);

    int r[8];
    ngram_row_indices(tokens + (b << 11), s, r);

    v4f acc = (v4f){0.f, 0.f, 0.f, 0.f};
#pragma unroll
    for (int k = 0; k < NGRAMS; ++k) {
        const v4f* row = (const v4f*)(tables + (size_t)r[k] * DIM);
        acc += __builtin_nontemporal_load(&row[lane]);
    }
    acc *= (1.0f / (float)NGRAMS);
    v4f* o = (v4f*)(out + (size_t)pos * DIM);
    __builtin_nontemporal_store(acc, &o[lane]);
}

extern "C" void kernel_launch(void* const* d_in, const int* in_sizes, int n_in,
                              void* d_out, int out_size, void* d_ws, size_t ws_size,
                              hipStream_t stream) {
    const int*   tokens = (const int*)d_in[0];
    const float* tables = (const float*)d_in[1];
    float*       out    = (float*)d_out;
    (void)in_sizes; (void)n_in; (void)out_size;

    const size_t ws_needed = (size_t)POSITIONS * 8 * sizeof(int);   // 2 MB
    if (d_ws != nullptr && ws_size >= ws_needed) {
        int* rws = (int*)d_ws;
        ngram_hash_kernel<<<dim3(POSITIONS / 256), 256, 0, stream>>>(tokens, rws);
        ngram_gather_kernel<<<dim3(POSITIONS / WAVES_PER_BLOCK), 256, 0, stream>>>(
            rws, tables, out);
    } else {
        ngram_fused_kernel<<<dim3(POSITIONS / WAVES_PER_BLOCK), 256, 0, stream>>>(
            tokens, tables, out);
    }
}